// LSTMPredictor_55611236548996
// MI455X (gfx1250) — compile-verified
//
#include <hip/hip_runtime.h>
#include <hip/hip_bf16.h>

#define H 1024
#define TSTEPS 2048
#define NWG 64         // persistent LSTM workgroups; each owns 16 h-elements (64 gate rows)

typedef __attribute__((ext_vector_type(2))) float v2f;
typedef __attribute__((ext_vector_type(8))) float v8f;

// ---------------------------------------------------------------------------
// prep: Wsum = Wih + Whh   (x==h for t>=1, so fold the two matmuls into one)
// ---------------------------------------------------------------------------
__global__ __launch_bounds__(256) void prep_kernel(const float* __restrict__ a,
                                                   const float* __restrict__ b,
                                                   float* __restrict__ c) {
    size_t i = (size_t)blockIdx.x * 256 + threadIdx.x;   // grid covers 4*H*H
    c[i] = a[i] + b[i];
}

// ---------------------------------------------------------------------------
// LSTM persistent kernel helpers
// ---------------------------------------------------------------------------
__device__ __forceinline__ void step_epilogue(float acc[8], float* gates_s,
        int tid, int wave, int lane, int base,
        float bs0, float bs1, float bs2, float bs3,
        float& c, int t, float* __restrict__ hbuf, float* __restrict__ hs,
        int* cnt, int& tgt)
{
    // cross-lane reduce each of the 8 row partial sums (wave32 butterfly)
    #pragma unroll
    for (int i = 0; i < 8; ++i) {
        #pragma unroll
        for (int off = 16; off > 0; off >>= 1)
            acc[i] += __shfl_xor(acc[i], off, 32);
    }
    // lane i (i<8) deposits row i of this wave into LDS (cndmask chain, no
    // dynamic register indexing)
    float v = acc[0];
    #pragma unroll
    for (int i = 1; i < 8; ++i) v = (lane == i) ? acc[i] : v;
    if (lane < 8) gates_s[wave * 8 + lane] = v;
    __syncthreads();

    // threads 0..15 own h-elements base..base+15: gate nonlinearity + state
    if (tid < 16) {
        float gi = gates_s[tid]          + bs0;   // i
        float gf = gates_s[16 + tid]     + bs1;   // f
        float gg = gates_s[32 + tid]     + bs2;   // g
        float go = gates_s[48 + tid]     + bs3;   // o
        float si = 1.f / (1.f + __expf(-gi));
        float sf = 1.f / (1.f + __expf(-gf));
        float so = 1.f / (1.f + __expf(-go));
        c = sf * c + si * tanhf(gg);
        float hn = so * tanhf(c);
        hbuf[((t + 1) & 1) * H + base + tid] = hn;      // next step reads this
        hs[(size_t)t * H + base + tid]       = hn;      // history for the MLP
    }
    __threadfence();
    __syncthreads();
    // device-wide split barrier: monotonic counter, one arrival per WG per step
    if (tid == 0) {
        __hip_atomic_fetch_add(cnt, 1, __ATOMIC_RELEASE, __HIP_MEMORY_SCOPE_AGENT);
        while (__hip_atomic_load(cnt, __ATOMIC_ACQUIRE, __HIP_MEMORY_SCOPE_AGENT) < tgt)
            __builtin_amdgcn_s_sleep(1);
    }
    tgt += NWG;
    __syncthreads();
}

// Persistent cooperative LSTM. 64 WGs x 256 threads. WG b owns h[b*16..b*16+15]
// i.e. gate rows {g*H + b*16 + e}. Each wave handles 8 rows; each lane keeps an
// 8x32 weight sub-tile of Wsum in VGPRs for the entire 2048-step loop. Each
// lane covers the blocked k-slice [lane*32, lane*32+32) so every per-step load
// is a b128 and each wave pulls one contiguous 4KB line of h.
__global__ __launch_bounds__(256) void lstm_kernel(
        const float* __restrict__ z,    const float* __restrict__ Wih,
        const float* __restrict__ bih,  const float* __restrict__ bhh,
        const float* __restrict__ Wsum, float* __restrict__ hbuf,
        float* __restrict__ hs,         int* cnt)
{
    __shared__ float gates_s[64];
    const int tid  = threadIdx.x;
    const int wave = tid >> 5, lane = tid & 31;
    const int base = blockIdx.x * 16;
    const int kofs = lane * 32;                 // this lane's k-slice

    float bs0 = 0.f, bs1 = 0.f, bs2 = 0.f, bs3 = 0.f, c = 0.f;
    if (tid < 16) {
        bs0 = bih[         base + tid] + bhh[         base + tid];
        bs1 = bih[    H +  base + tid] + bhh[    H +  base + tid];
        bs2 = bih[2 * H +  base + tid] + bhh[2 * H +  base + tid];
        bs3 = bih[3 * H +  base + tid] + bhh[3 * H +  base + tid];
    }
    int tgt = NWG;

    // ---- step 0 (peeled): gates = z @ Wih.T (+biases), h == 0 -> stream Wih
    {
        float xr[32];
        const float4* zv = (const float4*)(z + kofs);
        #pragma unroll
        for (int j = 0; j < 8; ++j) {
            float4 t4 = zv[j];
            xr[4*j+0] = t4.x; xr[4*j+1] = t4.y; xr[4*j+2] = t4.z; xr[4*j+3] = t4.w;
        }
        float acc[8];
        #pragma unroll
        for (int i = 0; i < 8; ++i) {
            int lr = wave * 8 + i;
            const float4* wrow = (const float4*)(Wih +
                (size_t)((lr >> 4) * H + base + (lr & 15)) * H + kofs);
            float s = 0.f;
            #pragma unroll
            for (int j = 0; j < 8; ++j) {
                float4 w = wrow[j];
                s += w.x * xr[4*j+0] + w.y * xr[4*j+1] + w.z * xr[4*j+2] + w.w * xr[4*j+3];
            }
            acc[i] = s;
        }
        step_epilogue(acc, gates_s, tid, wave, lane, base, bs0, bs1, bs2, bs3,
                      c, 0, hbuf, hs, cnt, tgt);
    }

    // ---- cache this lane's 8x32 tile of Wsum in VGPRs (256 regs/lane) ----
    float ws[8][32];
    #pragma unroll
    for (int i = 0; i < 8; ++i) {
        int lr = wave * 8 + i;
        const float4* wrow = (const float4*)(Wsum +
            (size_t)((lr >> 4) * H + base + (lr & 15)) * H + kofs);
        #pragma unroll
        for (int j = 0; j < 8; ++j) {
            float4 w = wrow[j];
            ws[i][4*j+0] = w.x; ws[i][4*j+1] = w.y; ws[i][4*j+2] = w.z; ws[i][4*j+3] = w.w;
        }
    }

    // ---- steps 1..T-1: gates = h @ Wsum.T, weights from registers ----
    for (int t = 1; t < TSTEPS; ++t) {
        const float4* hv = (const float4*)(hbuf + (t & 1) * H + kofs);
        float hx[32];
        #pragma unroll
        for (int j = 0; j < 8; ++j) {
            float4 t4 = hv[j];
            hx[4*j+0] = t4.x; hx[4*j+1] = t4.y; hx[4*j+2] = t4.z; hx[4*j+3] = t4.w;
        }
        float acc[8];
        #pragma unroll
        for (int i = 0; i < 8; ++i) {
            float s = 0.f;
            #pragma unroll
            for (int j = 0; j < 32; ++j) s += ws[i][j] * hx[j];
            acc[i] = s;
        }
        step_epilogue(acc, gates_s, tid, wave, lane, base, bs0, bs1, bs2, bs3,
                      c, t, hbuf, hs, cnt, tgt);
    }
}

// ---------------------------------------------------------------------------
// MLP layer 1 via fp32 WMMA: a1 = relu(hs @ W1.T + b1), (2048x1024)x(1024x32)
// One wave per 16-row M-tile; two 16-col N accumulators; K swept by
// V_WMMA_F32_16X16X4_F32 (ISA 7.12.2 operand layouts).
// ---------------------------------------------------------------------------
__global__ __launch_bounds__(256) void mlp1_kernel(const float* __restrict__ hs,
        const float* __restrict__ W1, const float* __restrict__ b1,
        float* __restrict__ a1)
{
    const int tid  = threadIdx.x;
    const int wave = tid >> 5, lane = tid & 31;
    const int m0   = (blockIdx.x * 8 + wave) * 16;     // 16 blocks x 8 waves = 128 tiles
    const int hi16 = lane >> 4, l15 = lane & 15;

    // A (16x4, M over lane halves, K even/odd interleaved across the 2 VGPRs):
    //   lane half h holds K = k+2h (VGPR0) and K = k+2h+1 (VGPR1) for row m0+l15
    const float* arow  = hs + (size_t)(m0 + l15) * H + 2 * hi16;
    // B (4x16 = K x N): mirrored layout, N = l15 (+16 for the second tile)
    const float* brow0 = W1 + (size_t)l15        * H + 2 * hi16;
    const float* brow1 = W1 + (size_t)(16 + l15) * H + 2 * hi16;

    v8f acc0 = {}, acc1 = {};
    #pragma unroll 4
    for (int k = 0; k < H; k += 4) {
        v2f a  = *(const v2f*)(arow  + k);
        v2f b0 = *(const v2f*)(brow0 + k);
        v2f bb = *(const v2f*)(brow1 + k);
        acc0 = __builtin_amdgcn_wmma_f32_16x16x4_f32(false, a, false, b0,
                                                     (short)0, acc0, false, false);
        acc1 = __builtin_amdgcn_wmma_f32_16x16x4_f32(false, a, false, bb,
                                                     (short)0, acc1, false, false);
    }
    const float bias0 = b1[l15], bias1 = b1[16 + l15];
    #pragma unroll
    for (int v = 0; v < 8; ++v) {           // D: VGPR v holds rows v / v+8
        int m = m0 + v + hi16 * 8;
        float x0 = acc0[v] + bias0;
        float x1 = acc1[v] + bias1;
        a1[m * 32 + l15]      = x0 > 0.f ? x0 : 0.f;
        a1[m * 32 + 16 + l15] = x1 > 0.f ? x1 : 0.f;
    }
}

// ---------------------------------------------------------------------------
// MLP layers 2+3: per-timestep thread. W2/W3 staged into LDS with the CDNA5
// async-copy path (GLOBAL_LOAD_ASYNC_TO_LDS_B128, ASYNCcnt-tracked DMA, no
// VGPR round trip), then compute from LDS.
// ---------------------------------------------------------------------------
__global__ __launch_bounds__(256) void mlp23_kernel(const float* __restrict__ a1,
        const float* __restrict__ W2, const float* __restrict__ b2,
        const float* __restrict__ W3, const float* __restrict__ b3,
        float* __restrict__ out)
{
    __shared__ __align__(16) float w2s[32 * 32];
    __shared__ __align__(16) float w3s[64 * 32];
    __shared__ float b2s[32], b3s[64];
    const int tid = threadIdx.x;
    const int i4  = tid * 4;

    // async-stage W2 (1024 f32): one b128 per thread
    {
        unsigned lds = (unsigned)(size_t)&w2s[i4];
        const float* g = W2 + i4;
        asm volatile("global_load_async_to_lds_b128 %0, %1, off"
                     :: "v"(lds), "v"(g) : "memory");
    }
    // async-stage W3 (2048 f32): two b128 per thread
    #pragma unroll
    for (int r = 0; r < 2; ++r) {
        unsigned lds = (unsigned)(size_t)&w3s[i4 + r * 1024];
        const float* g = W3 + i4 + r * 1024;
        asm volatile("global_load_async_to_lds_b128 %0, %1, off"
                     :: "v"(lds), "v"(g) : "memory");
    }
    if (tid < 32) b2s[tid] = b2[tid];
    if (tid < 64) b3s[tid] = b3[tid];
    asm volatile("s_wait_asynccnt 0x0" ::: "memory");
    __syncthreads();

    const int t = blockIdx.x * 256 + tid;   // 8 blocks x 256 = 2048 timesteps
    float x[32];
    const float4* av = (const float4*)(a1 + (size_t)t * 32);
    #pragma unroll
    for (int i = 0; i < 8; ++i) {
        float4 v = av[i];
        x[4*i+0] = v.x; x[4*i+1] = v.y; x[4*i+2] = v.z; x[4*i+3] = v.w;
    }
    float y[32];
    #pragma unroll 4
    for (int j = 0; j < 32; ++j) {
        float s = b2s[j];
        #pragma unroll
        for (int k = 0; k < 32; ++k) s += w2s[j * 32 + k] * x[k];
        y[j] = s > 0.f ? s : 0.f;
    }
    float4* ov = (float4*)(out + (size_t)t * 64);
    #pragma unroll 2
    for (int j = 0; j < 64; j += 4) {
        float4 o;
        float* op = (float*)&o;
        #pragma unroll
        for (int q = 0; q < 4; ++q) {
            float s = b3s[j + q];
            #pragma unroll
            for (int k = 0; k < 32; ++k) s += w3s[(j + q) * 32 + k] * y[k];
            op[q] = s;
        }
        ov[j >> 2] = o;
    }
}

// ---------------------------------------------------------------------------
// launch
// ---------------------------------------------------------------------------
extern "C" void kernel_launch(void* const* d_in, const int* in_sizes, int n_in,
                              void* d_out, int out_size, void* d_ws, size_t ws_size,
                              hipStream_t stream) {
    const float* z    = (const float*)d_in[0];
    const float* Wih  = (const float*)d_in[1];
    const float* Whh  = (const float*)d_in[2];
    const float* b_ih = (const float*)d_in[3];
    const float* b_hh = (const float*)d_in[4];
    const float* W1   = (const float*)d_in[5];
    const float* b1   = (const float*)d_in[6];
    const float* W2   = (const float*)d_in[7];
    const float* b2   = (const float*)d_in[8];
    const float* W3   = (const float*)d_in[9];
    const float* b3   = (const float*)d_in[10];
    float* out = (float*)d_out;

    // workspace layout (~25 MB):
    //   [0,4K)        barrier counter (re-zeroed every launch)
    //   [4K,12K)      h double-buffer (2 x 1024 f32)
    //   [16K,272K)    a1 (2048 x 32 f32)
    //   [1M,17M)      Wsum (4096 x 1024 f32)
    //   [17M,25M)     hs  (2048 x 1024 f32)
    char*  ws   = (char*)d_ws;
    int*   cnt  = (int*)  (ws);
    float* hbuf = (float*)(ws + 4096);
    float* a1   = (float*)(ws + (16u << 10));
    float* Wsum = (float*)(ws + (1u  << 20));
    float* hs   = (float*)(ws + (17u << 20));

    hipMemsetAsync(cnt, 0, 4096, stream);

    prep_kernel <<<(4 * H * H) / 256, 256, 0, stream>>>(Wih, Whh, Wsum);
    lstm_kernel <<<NWG, 256, 0, stream>>>(z, Wih, b_ih, b_hh, Wsum, hbuf, hs, cnt);
    mlp1_kernel <<<16, 256, 0, stream>>>(hs, W1, b1, a1);
    mlp23_kernel<<<8, 256, 0, stream>>>(a1, W2, b2, W3, b3, out);
}